// SupConLoss_89670327206311
// MI455X (gfx1250) — compile-verified
//
#include <hip/hip_runtime.h>
#include <hip/hip_bf16.h>

typedef float v2f __attribute__((ext_vector_type(2)));
typedef float v8f __attribute__((ext_vector_type(8)));

#define N_TOTAL 8192
#define DIM 256
#define RB 64            // rows per block
#define CB 64            // cols per inner iteration (double-buffered)
#define NCHUNK 4
#define CHUNK_COLS 2048  // 8192 / NCHUNK
#define NIT (CHUNK_COLS / CB)  // 32 iterations
#define SA 260           // padded LDS row stride in floats (4-bank advance/row)
#define TPB 256

__device__ __forceinline__ v8f wmma_f32(v2f a, v2f b, v8f c) {
  // V_WMMA_F32_16X16X4_F32: D = A(16x4) x B(4x16) + C(16x16)
  return __builtin_amdgcn_wmma_f32_16x16x4_f32(
      /*neg_a=*/false, a, /*neg_b=*/false, b,
      /*c_mod=*/(short)0, c, /*reuse_a=*/false, /*reuse_b=*/false);
}

// Async DMA: global -> LDS, 16B per lane, tracked by ASYNCcnt (no VGPR staging).
__device__ __forceinline__ void async_ld128(unsigned int lds_addr, unsigned int goff,
                                            const float* base) {
  asm volatile("global_load_async_to_lds_b128 %0, %1, %2"
               :: "v"(lds_addr), "v"(goff), "s"(base)
               : "memory");
}
__device__ __forceinline__ void wait_async0() {
  asm volatile("s_wait_asynccnt 0" ::: "memory");
}
__device__ __forceinline__ unsigned int lds_off(const void* p) {
  // generic shared pointer: low 32 bits == wave-relative LDS byte address
  return (unsigned int)(unsigned long long)p;
}
// Opaque value copy: the whole large address component goes through this so the
// compiler cannot re-associate any big constant into per-load offsets; only the
// small k-step immediates remain on top -> ds_load_2addr_b64 merging works.
__device__ __forceinline__ int opaque_i(int x) {
  asm volatile("" : "+v"(x));
  return x;
}

// Issue async loads for a tile of `rows` contrast rows starting at global row g0
// into LDS at byte address ldsBase (row stride SA floats). 64 float4 per row.
__device__ __forceinline__ void issue_tile_async(const float* feat, unsigned int ldsBase,
                                                 int rows, int g0, int tid) {
#pragma unroll 4
  for (int v = tid; v < rows * (DIM / 4); v += TPB) {
    int row = v >> 6;  // DIM/4 == 64
    int f4 = v & 63;
    int g = g0 + row;  // contrast row -> features[(g%4096), g/4096, :]
    unsigned int goff =
        (unsigned int)(((g & 4095) * 2 + (g >> 12)) * DIM + f4 * 4) * 4u;
    unsigned int laddr = ldsBase + (unsigned int)(row * SA + f4 * 4) * 4u;
    async_ld128(laddr, goff, feat);
  }
}

// Fused tiled Gram-matrix GEMM + online softmax row statistics.
// Block (bx, by): rows [bx*64, bx*64+64) x cols [by*2048, by*2048+2048).
__global__ __launch_bounds__(TPB) void supcon_gemm_stats(
    const float* __restrict__ feat, const long long* __restrict__ labels,
    float4* __restrict__ partials) {
  extern __shared__ float smem[];
  float* As = smem;                         // RB x SA (LDS offset 0)
  float* Bs0 = smem + RB * SA;              // CB x SA (buffer 0); buffer 1 follows
  float4* stats = (float4*)(Bs0 + 2 * CB * SA);  // [2 col-wave-groups][RB]

  const int tid = threadIdx.x;
  const int lane = tid & 31;
  const int wave = tid >> 5;
  const int wr = wave >> 1;   // row group (0..3): rows 16*wr..
  const int wcg = wave & 1;   // col group (0..1): cols 32*wcg..
  const int m16 = lane & 15;
  const int h = lane >> 4;    // lane half (rows +8)

  const int R0g = blockIdx.x * RB;
  const int chunk = blockIdx.y;

  const unsigned int asBase = lds_off(As);
  const unsigned int bsBase = lds_off(Bs0);

  // ---- Prologue: async-stage A tile and first B tile ----
  issue_tile_async(feat, asBase, RB, R0g, tid);
  issue_tile_async(feat, bsBase, CB, chunk * CHUNK_COLS, tid);

  // Row labels for this wave's 16 rows (fixed across all column tiles)
  int labi[8];
#pragma unroll
  for (int r = 0; r < 8; ++r)
    labi[r] = (int)labels[(R0g + 16 * wr + 8 * h + r) & 4095];

  // Online per-row-slot stats (rows in (r, lane-half); cols in lanes)
  float mS[8], sS[8], pS[8], cS[8];
#pragma unroll
  for (int r = 0; r < 8; ++r) {
    mS[r] = -1.0e30f; sS[r] = 0.0f; pS[r] = 0.0f; cS[r] = 0.0f;
  }

  // A fragments: base at LDS offset 0, per-lane row offset in the base VGPR,
  // small k immediates on top -> already merges into ds_load_2addr_b64.
  const float* pa = As + (16 * wr + m16) * SA;
  // B fragments: the full large element offset (static Bs0 offset + per-wave
  // row offset + ping-pong buffer offset) is laundered per iteration below.
  const int rowOffB = RB * SA + (32 * wcg + m16) * SA;  // element offset from smem
  const int koff = 2 * h;

  for (int it = 0; it < NIT; ++it) {
    const int bufOff = (it & 1) * (CB * SA);
    const int C0 = chunk * CHUNK_COLS + it * CB;

    wait_async0();    // this thread's async copies for tile `it` (and earlier) done
    __syncthreads();  // all threads' copies done AND compute on the other buffer done

    if (it + 1 < NIT)  // overlap: DMA next tile while computing this one
      issue_tile_async(feat, bsBase + (unsigned int)(((it + 1) & 1) * (CB * SA)) * 4u,
                       CB, C0 + CB, tid);

    // Dedicated opaque base registers: nothing big left to fold per-load.
    const float* pb0 = smem + opaque_i(rowOffB + bufOff);
    const float* pb1 = smem + opaque_i(rowOffB + bufOff + 16 * SA);

    // ---- 16x32 wave tile: 2 accumulators, K = 256 in steps of 4 ----
    v8f acc0 = {}, acc1 = {};
#pragma unroll 16
    for (int kk = 0; kk < DIM; kk += 4) {
      int o = kk + koff;
      v2f a  = *(const v2f*)(pa + o);
      v2f b0 = *(const v2f*)(pb0 + o);
      v2f b1 = *(const v2f*)(pb1 + o);
      acc0 = wmma_f32(a, b0, acc0);
      acc1 = wmma_f32(a, b1, acc1);
    }

    // ---- Online softmax / mask epilogue on the 16x32 logits ----
    const int ig_base = R0g + 16 * wr + 8 * h;  // row = ig_base + r
#pragma unroll
    for (int sj = 0; sj < 2; ++sj) {
      int jg = C0 + 32 * wcg + 16 * sj + m16;  // this lane's column
      int labj = (int)labels[jg & 4095];
      v8f acc = (sj == 0) ? acc0 : acc1;
#pragma unroll
      for (int r = 0; r < 8; ++r) {
        float e = acc[r] * 10.0f;  // logits = dot / T, T = 0.1
        bool diag = (ig_base + r) == jg;
        bool pos = (!diag) && (labi[r] == labj);
        float mo = mS[r];
        float mn = fmaxf(mo, e);                    // max includes diagonal (ref)
        float sadd = diag ? 0.0f : __expf(e - mn);  // diag excluded from expsum
        sS[r] = sS[r] * __expf(mo - mn) + sadd;
        mS[r] = mn;
        pS[r] += pos ? e : 0.0f;
        cS[r] += pos ? 1.0f : 0.0f;
      }
    }
  }

  // ---- Reduce across the 16 lanes of each half (each half holds distinct rows) ----
#pragma unroll
  for (int r = 0; r < 8; ++r) {
    float m = mS[r], s = sS[r], p = pS[r], c = cS[r];
#pragma unroll
    for (int off = 1; off < 16; off <<= 1) {
      float mo = __shfl_xor(m, off, 16);
      float so = __shfl_xor(s, off, 16);
      float mn = fmaxf(m, mo);
      s = s * __expf(m - mn) + so * __expf(mo - mn);
      m = mn;
      p += __shfl_xor(p, off, 16);
      c += __shfl_xor(c, off, 16);
    }
    if (m16 == 0) {
      int rowLocal = 16 * wr + 8 * h + r;
      stats[wcg * RB + rowLocal] = make_float4(m, s, p, c);
    }
  }
  __syncthreads();

  // ---- Merge the 2 column-wave groups, emit per-(row, chunk) partial ----
  if (tid < RB) {
    float M = -1.0e30f, S = 0.0f, P = 0.0f, C = 0.0f;
#pragma unroll
    for (int w = 0; w < 2; ++w) {
      float4 q = stats[w * RB + tid];
      float Mn = fmaxf(M, q.x);
      S = S * __expf(M - Mn) + q.y * __expf(q.x - Mn);
      M = Mn; P += q.z; C += q.w;
    }
    partials[chunk * N_TOTAL + R0g + tid] = make_float4(M, S, P, C);
  }
}

// Merge the NCHUNK column-chunk partials per row -> per-row loss.
__global__ __launch_bounds__(TPB) void supcon_row_loss(
    const float4* __restrict__ partials, float* __restrict__ loss) {
  int i = blockIdx.x * TPB + threadIdx.x;
  float M = -1.0e30f, S = 0.0f, P = 0.0f, C = 0.0f;
#pragma unroll
  for (int k = 0; k < NCHUNK; ++k) {
    float4 q = partials[k * N_TOTAL + i];
    float Mn = fmaxf(M, q.x);
    S = S * __expf(M - Mn) + q.y * __expf(q.x - Mn);
    M = Mn; P += q.z; C += q.w;
  }
  // mean_log_prob_pos = (P - C*M)/C - log(S);  loss_i = -(T/baseT) * that
  float lp = (P - C * M) / C - logf(S);
  loss[i] = -(0.1f / 0.07f) * lp;
}

// Deterministic final mean (fixed summation order; no float atomics).
__global__ __launch_bounds__(TPB) void supcon_reduce(
    const float* __restrict__ loss, float* __restrict__ out) {
  __shared__ float red[TPB];
  int tid = threadIdx.x;
  float s = 0.0f;
  for (int i = tid; i < N_TOTAL; i += TPB) s += loss[i];
  red[tid] = s;
  __syncthreads();
  for (int w = TPB / 2; w > 0; w >>= 1) {
    if (tid < w) red[tid] += red[tid + w];
    __syncthreads();
  }
  if (tid == 0) out[0] = red[0] * (1.0f / (float)N_TOTAL);
}

extern "C" void kernel_launch(void* const* d_in, const int* in_sizes, int n_in,
                              void* d_out, int out_size, void* d_ws, size_t ws_size,
                              hipStream_t stream) {
  const float* feat = (const float*)d_in[0];           // (4096, 2, 256) f32
  const long long* labels = (const long long*)d_in[1]; // (4096, 1) i64
  float* out = (float*)d_out;

  float4* partials = (float4*)d_ws;  // NCHUNK * 8192 float4 = 512 KB
  float* lossArr = (float*)((char*)d_ws + (size_t)NCHUNK * N_TOTAL * sizeof(float4));

  size_t shmem = (size_t)(RB + 2 * CB) * SA * sizeof(float) + 2 * RB * sizeof(float4);

  dim3 grid1(N_TOTAL / RB, NCHUNK);  // 128 x 4
  supcon_gemm_stats<<<grid1, TPB, shmem, stream>>>(feat, labels, partials);
  supcon_row_loss<<<N_TOTAL / TPB, TPB, 0, stream>>>(partials, lossArr);
  supcon_reduce<<<1, TPB, 0, stream>>>(lossArr, out);
}